// BaselineGAT_41764261987077
// MI455X (gfx1250) — compile-verified
//
#include <hip/hip_runtime.h>
#include <hip/hip_bf16.h>
#include <math.h>

// GAT (2 layers, H=4, C=128) + classifier for MI455X (gfx1250, wave32).
// GEMMs via V_WMMA_F32_16X16X4_F32; segment softmax via encoded-uint atomicMax
// + f32 atomicAdd; aggregation = wave-per-edge scatter with f32 atomics.

#define NEG_SLOPE 0.2f
#define EPS_DEN 1e-16f

typedef float v2f __attribute__((ext_vector_type(2)));
typedef float v8f __attribute__((ext_vector_type(8)));

// Order-preserving float<->uint mapping for atomicMax on floats.
__device__ __forceinline__ unsigned fenc(float f) {
  unsigned u = __float_as_uint(f);
  return (u & 0x80000000u) ? ~u : (u | 0x80000000u);
}
__device__ __forceinline__ float fdec(unsigned u) {
  unsigned b = (u & 0x80000000u) ? (u ^ 0x80000000u) : ~u;
  return __uint_as_float(b);
}

// ---------------------------------------------------------------------------
// GEMM: C[M,N] = A[M,K] @ B[K,N] (+bias, +relu). Row-major. M%16==0, K%4==0,
// N % 64 == 0. Block = 128 threads = 4 waves; wave handles one 16-row tile and
// 4 consecutive 16-col tiles (A fragment reused 4x per k-step).
// EPI: 0 = none, 2 = bias + relu.
// ---------------------------------------------------------------------------
template <int EPI>
__global__ void gemm_wmma_f32(const float* __restrict__ A,
                              const float* __restrict__ B,
                              const float* __restrict__ bias,
                              float* __restrict__ C, int M, int N, int K) {
  constexpr int NT = 4;
  const int lane = threadIdx.x & 31;
  const int wave = threadIdx.x >> 5;
  const int l16 = lane & 15;
  const int half = lane >> 4;  // 0: K={0,1}/rows M..M+7 ; 1: K={2,3}/rows M+8..
  const int mtile = blockIdx.y * 4 + wave;
  if (mtile * 16 >= M) return;  // wave-uniform
  const int nbase = blockIdx.x * (16 * NT);

  const float* __restrict__ arow = A + (size_t)(mtile * 16 + l16) * K;

  v8f acc[NT] = {};
  for (int k = 0; k < K; k += 4) {
    const int kk = k + half * 2;
    // A 16x4 f32 fragment: lane holds A[M = lane&15, kk..kk+1] (contig float2)
    v2f a = *(const v2f*)(arow + kk);
#pragma unroll
    for (int t = 0; t < NT; ++t) {
      const int col = nbase + t * 16 + l16;
      // B 4x16 f32 fragment: lane holds B[kk..kk+1, N = lane&15]
      v2f b;
      b.x = B[(size_t)(kk + 0) * N + col];
      b.y = B[(size_t)(kk + 1) * N + col];
      acc[t] = __builtin_amdgcn_wmma_f32_16x16x4_f32(
          false, a, false, b, (short)0, acc[t], false, false);
    }
  }

#pragma unroll
  for (int t = 0; t < NT; ++t) {
    const int col = nbase + t * 16 + l16;
    float bv = (EPI > 0) ? bias[col] : 0.0f;
#pragma unroll
    for (int r = 0; r < 8; ++r) {
      const int orow = mtile * 16 + half * 8 + r;  // C/D VGPR layout
      float v = acc[t][r];
      if (EPI > 0) v += bv;
      if (EPI == 2) v = fmaxf(v, 0.0f);
      C[(size_t)orow * N + col] = v;
    }
  }
}

// ---------------------------------------------------------------------------
// alpha_{src,dst}[n,h] = sum_c h[n,h,c] * a_{src,dst}[h,c]   (C=128, H=4)
// One wave per (n,h): lane holds float4, xor-shuffle reduction.
// ---------------------------------------------------------------------------
__global__ void alpha_kernel(const float* __restrict__ h,
                             const float* __restrict__ a_src,
                             const float* __restrict__ a_dst,
                             float* __restrict__ as, float* __restrict__ ad,
                             int NH) {
  const int gw = (int)((blockIdx.x * blockDim.x + threadIdx.x) >> 5);
  const int lane = threadIdx.x & 31;
  if (gw >= NH) return;  // wave-uniform
  const int head = gw & 3;
  const float4 hv = ((const float4*)(h + (size_t)gw * 128))[lane];
  const float4 sv = ((const float4*)(a_src + head * 128))[lane];
  const float4 dv = ((const float4*)(a_dst + head * 128))[lane];
  float s = hv.x * sv.x + hv.y * sv.y + hv.z * sv.z + hv.w * sv.w;
  float d = hv.x * dv.x + hv.y * dv.y + hv.z * dv.z + hv.w * dv.w;
#pragma unroll
  for (int off = 16; off > 0; off >>= 1) {
    s += __shfl_xor(s, off, 32);
    d += __shfl_xor(d, off, 32);
  }
  if (lane == 0) {
    as[gw] = s;
    ad[gw] = d;
  }
}

__global__ void init_seg(unsigned* __restrict__ seg_m, float* __restrict__ seg_s,
                         int n) {
  int i = blockIdx.x * blockDim.x + threadIdx.x;
  if (i < n) {
    seg_m[i] = 0u;  // fenc(x) > 0 for all finite x => 0 acts as -inf
    seg_s[i] = 0.0f;
  }
}

__global__ void init_out_bias(float* __restrict__ out,
                              const float* __restrict__ bias, size_t total,
                              int HC) {
  size_t i = (size_t)blockIdx.x * blockDim.x + threadIdx.x;
  if (i < total) out[i] = bias[i & (HC - 1)];  // HC = 512 power of two
}

// Pass 1: e = leakyrelu(alpha_src[src] + alpha_dst[dst]); segment max into m.
__global__ void edge_max(const int* __restrict__ ei, int E, int Etot,
                         const float* __restrict__ as,
                         const float* __restrict__ ad, float* __restrict__ ebuf,
                         unsigned* __restrict__ seg_m) {
  int e = blockIdx.x * blockDim.x + threadIdx.x;
  if (e >= Etot) return;
  int s, d;
  if (e < E) {
    s = ei[e];
    d = ei[E + e];
  } else {
    s = d = e - E;  // self loop
  }
#pragma unroll
  for (int h = 0; h < 4; ++h) {
    float v = as[s * 4 + h] + ad[d * 4 + h];
    v = v > 0.0f ? v : NEG_SLOPE * v;
    ebuf[(size_t)e * 4 + h] = v;
    atomicMax(&seg_m[d * 4 + h], fenc(v));
  }
}

// Pass 2: p = exp(e - m[dst]); segment sum into s; overwrite ebuf with p.
__global__ void edge_expsum(const int* __restrict__ ei, int E, int Etot,
                            const unsigned* __restrict__ seg_m,
                            float* __restrict__ ebuf,
                            float* __restrict__ seg_s) {
  int e = blockIdx.x * blockDim.x + threadIdx.x;
  if (e >= Etot) return;
  int d = (e < E) ? ei[E + e] : (e - E);
#pragma unroll
  for (int h = 0; h < 4; ++h) {
    float m = fdec(seg_m[d * 4 + h]);
    float p = expf(ebuf[(size_t)e * 4 + h] - m);
    ebuf[(size_t)e * 4 + h] = p;
    atomicAdd(&seg_s[d * 4 + h], p);
  }
}

// Pass 3: out[dst] += (p / (s[dst]+eps)) * h[src]. One wave per edge; lane
// handles one float4 per head (4 heads x 128 channels).
__global__ void edge_aggregate(const int* __restrict__ ei, int E, int Etot,
                               const float* __restrict__ ebuf,
                               const float* __restrict__ seg_s,
                               const float* __restrict__ hfeat,
                               float* __restrict__ out) {
  const int gw = (int)((blockIdx.x * blockDim.x + threadIdx.x) >> 5);
  const int lane = threadIdx.x & 31;
  if (gw >= Etot) return;  // wave-uniform
  int s, d;
  if (gw < E) {
    s = ei[gw];
    d = ei[E + gw];
  } else {
    s = d = gw - E;
  }
  const float4* __restrict__ hp = (const float4*)(hfeat + (size_t)s * 512);
#pragma unroll
  for (int h = 0; h < 4; ++h) {
    const float attn =
        ebuf[(size_t)gw * 4 + h] / (seg_s[d * 4 + h] + EPS_DEN);
    const float4 hv = hp[h * 32 + lane];
    float* ob = out + (size_t)d * 512 + h * 128 + lane * 4;
    atomicAdd(ob + 0, attn * hv.x);
    atomicAdd(ob + 1, attn * hv.y);
    atomicAdd(ob + 2, attn * hv.z);
    atomicAdd(ob + 3, attn * hv.w);
  }
}

__global__ void elu_kernel(float* __restrict__ x, size_t n) {
  size_t i = (size_t)blockIdx.x * blockDim.x + threadIdx.x;
  if (i < n) {
    float v = x[i];
    x[i] = v > 0.0f ? v : expm1f(v);
  }
}

// logits[N,10] = z[N,128] @ Wc2[128,10] + bc2
__global__ void final_logits(const float* __restrict__ z,
                             const float* __restrict__ Wc2,
                             const float* __restrict__ bc2,
                             float* __restrict__ out, int N) {
  int idx = blockIdx.x * blockDim.x + threadIdx.x;
  int n = idx >> 4;
  int j = idx & 15;
  if (n >= N || j >= 10) return;
  const float* zp = z + (size_t)n * 128;
  float acc = bc2[j];
#pragma unroll 8
  for (int k = 0; k < 128; ++k) acc = fmaf(zp[k], Wc2[k * 10 + j], acc);
  out[(size_t)n * 10 + j] = acc;
}

// ---------------------------------------------------------------------------

extern "C" void kernel_launch(void* const* d_in, const int* in_sizes, int n_in,
                              void* d_out, int out_size, void* d_ws,
                              size_t ws_size, hipStream_t stream) {
  const float* x   = (const float*)d_in[0];
  const int*   ei  = (const int*)d_in[1];
  const float* W1  = (const float*)d_in[2];
  const float* as1 = (const float*)d_in[3];
  const float* ad1 = (const float*)d_in[4];
  const float* b1  = (const float*)d_in[5];
  const float* W2  = (const float*)d_in[6];
  const float* as2 = (const float*)d_in[7];
  const float* ad2 = (const float*)d_in[8];
  const float* b2  = (const float*)d_in[9];
  const float* Wc1 = (const float*)d_in[10];
  const float* bc1 = (const float*)d_in[11];
  const float* Wc2 = (const float*)d_in[12];
  const float* bc2 = (const float*)d_in[13];
  float* out = (float*)d_out;

  const int F_in = 336, H = 4, C = 128, HC = 512;
  const int Nn = in_sizes[0] / F_in;   // 50000
  const int E = in_sizes[1] / 2;       // 500000
  const int Etot = E + Nn;             // with self loops
  const int NH = Nn * H;

  // Workspace layout (~242 MB)
  float* p = (float*)d_ws;
  float* buf_h = p;            p += (size_t)Nn * HC;   // transformed features
  float* buf_o = p;            p += (size_t)Nn * HC;   // aggregated output
  float* alpha_s = p;          p += (size_t)NH;
  float* alpha_d = p;          p += (size_t)NH;
  unsigned* seg_m = (unsigned*)p; p += (size_t)NH;
  float* seg_s = p;            p += (size_t)NH;
  float* ebuf = p;             p += (size_t)Etot * H;
  float* buf_z = p;            p += (size_t)Nn * C;

  const dim3 gblk(128);
  const dim3 g512(HC / 64, (Nn / 16 + 3) / 4);  // N=512 GEMMs
  const dim3 g128(C / 64, (Nn / 16 + 3) / 4);   // N=128 GEMM

  auto gat_layer = [&](const float* feat_in, int K, const float* W,
                       const float* a_s, const float* a_d, const float* b) {
    gemm_wmma_f32<0><<<g512, gblk, 0, stream>>>(feat_in, W, nullptr, buf_h, Nn,
                                                HC, K);
    alpha_kernel<<<(NH * 32 + 127) / 128, 128, 0, stream>>>(buf_h, a_s, a_d,
                                                            alpha_s, alpha_d,
                                                            NH);
    init_seg<<<(NH + 255) / 256, 256, 0, stream>>>(seg_m, seg_s, NH);
    init_out_bias<<<(unsigned)(((size_t)Nn * HC + 255) / 256), 256, 0,
                    stream>>>(buf_o, b, (size_t)Nn * HC, HC);
    edge_max<<<(Etot + 255) / 256, 256, 0, stream>>>(ei, E, Etot, alpha_s,
                                                     alpha_d, ebuf, seg_m);
    edge_expsum<<<(Etot + 255) / 256, 256, 0, stream>>>(ei, E, Etot, seg_m,
                                                        ebuf, seg_s);
    edge_aggregate<<<(unsigned)(((size_t)Etot * 32 + 255) / 256), 256, 0,
                     stream>>>(ei, E, Etot, ebuf, seg_s, buf_h, buf_o);
  };

  // Layer 1
  gat_layer(x, F_in, W1, as1, ad1, b1);
  // ELU between layers (in place on buf_o)
  elu_kernel<<<(unsigned)(((size_t)Nn * HC + 255) / 256), 256, 0, stream>>>(
      buf_o, (size_t)Nn * HC);
  // Layer 2 (reads buf_o into buf_h before buf_o is re-initialized)
  gat_layer(buf_o, HC, W2, as2, ad2, b2);
  // Classifier: z = relu(h2 @ Wc1 + bc1)
  gemm_wmma_f32<2><<<g128, gblk, 0, stream>>>(buf_o, Wc1, bc1, buf_z, Nn, C,
                                              HC);
  // logits = z @ Wc2 + bc2
  final_logits<<<(unsigned)(((size_t)Nn * 16 + 255) / 256), 256, 0, stream>>>(
      buf_z, Wc2, bc2, out, Nn);
}